// RoundBoxSDF_76828374991337
// MI455X (gfx1250) — compile-verified
//
#include <hip/hip_runtime.h>

// RoundBox SDF + exponential smooth-min, n=32 primitives, P points.
// Strategy (MI455X / gfx1250, wave32):
//  - Affine part (T*p - c) done on the matrix unit with V_WMMA_F32_16X16X4_F32:
//      A[row, k] = k<3 ? (tfs+I)[prim][comp][k] : -centers[prim][comp]
//      B[k, pt]  = (px, py, pz, 1)
//    Rows padded to 4 per primitive -> 16 rows = 4 primitives per WMMA,
//    8 WMMAs = all 32 primitives for a 16-point tile.
//  - D layout (16x16 f32 C/D): lane holds point (lane&15), VGPR v holds row
//    v + 8*(lane>=16). With 4-row padding each lane gets TWO whole primitives
//    (d[0..2], d[4..6]) -> nonlinear SDF math stays lane-local f32 VALU.
//  - Per-point partials live in lanes l and l^16; combined with ds_swizzle
//    SWAPX16 before the final -log2(sum)*ln2/K.

typedef float v2f __attribute__((ext_vector_type(2)));
typedef float v8f __attribute__((ext_vector_type(8)));

#define SDF_EPS       1e-7f
#define SDF_LOGCLAMP  1e-6f
// exp(-16*sd) = exp2(sd * EXP2_SCALE)
#define EXP2_SCALE   (-23.083120654223414f)   // -16 * log2(e)
// -log(x)/16 = -log2(x) * LOG_SCALE
#define LOG_SCALE    (0.043321698784996581f)  // ln(2)/16

__global__ __launch_bounds__(256)
void roundbox_sdf_wmma(const float* __restrict__ p,
                       const float* __restrict__ centers,
                       const float* __restrict__ bhalf,
                       const float* __restrict__ tfs,
                       float* __restrict__ out,
                       int P, int numTiles)
{
  const int lane = threadIdx.x & 31;
  const bool hi  = lane >= 16;
  const int  m   = lane & 15;          // A-row / D-point index for this lane

  // ---------------- Build A (uniform per wave, reused across tiles) --------
  // A-matrix 16x4 layout: lane m holds row m: K0,K1 in VGPR0,1 (lanes 0-15);
  // K2,K3 in VGPR0,1 of lanes 16-31.
  float A0[8], A1[8];
  // Box half-extents for this lane's two primitives per WMMA a.
  float bb0[8], bb1[8], bb2[8], bb3[8], bb4[8], bb5[8];

#pragma unroll
  for (int a = 0; a < 8; ++a) {
    const int r    = a * 16 + m;       // global padded row (0..127)
    const int prim = r >> 2;           // 4 rows per primitive
    const int comp = r & 3;            // 0,1,2 = x,y,z ; 3 = pad
    float v0 = 0.0f, v1 = 0.0f;
    if (comp < 3) {
      const int k0 = hi ? 2 : 0;
      v0 = tfs[prim * 9 + comp * 3 + k0] + (comp == k0 ? 1.0f : 0.0f);
      v1 = hi ? -centers[prim * 3 + comp]
              :  tfs[prim * 9 + comp * 3 + 1] + (comp == 1 ? 1.0f : 0.0f);
    }
    A0[a] = v0;
    A1[a] = v1;

    // This lane's two primitives in WMMA a: rows {v,v+8*hi} -> prims pA,pA+1
    const int pA = a * 4 + (hi ? 2 : 0);
    bb0[a] = bhalf[pA * 3 + 0];
    bb1[a] = bhalf[pA * 3 + 1];
    bb2[a] = bhalf[pA * 3 + 2];
    bb3[a] = bhalf[pA * 3 + 3];   // == b[pA+1][0]
    bb4[a] = bhalf[pA * 3 + 4];
    bb5[a] = bhalf[pA * 3 + 5];
  }

  const int wavesPerBlock = blockDim.x >> 5;
  const int wave   = blockIdx.x * wavesPerBlock + (threadIdx.x >> 5);
  const int nWaves = gridDim.x * wavesPerBlock;

  // ---------------- Grid-stride loop over 16-point tiles --------------------
  for (int t = wave; t < numTiles; t += nWaves) {
    const int base = t * 16;
    int pidx = base + m;
    if (pidx >= P) pidx = P - 1;       // clamp (full EXEC kept for WMMA)

    // B-matrix 4x16: VGPR0 = K0 (lanes 0-15) / K2 (lanes 16-31),
    //                VGPR1 = K1 (lanes 0-15) / K3==1.0 (lanes 16-31)
    const float B0 = p[pidx * 3 + (hi ? 2 : 0)];
    const float B1 = hi ? 1.0f : p[pidx * 3 + 1];
    const v2f Bv = {B0, B1};

    float acc = 0.0f;

#pragma unroll
    for (int a = 0; a < 8; ++a) {
      const v2f Av = {A0[a], A1[a]};
      v8f c = {};
      // D = A*B + 0 ; D already equals (tp - center) per row.
      v8f d = __builtin_amdgcn_wmma_f32_16x16x4_f32(
          false, Av, false, Bv, (short)0, c, false, false);

      // ---- primitive #1 of this lane (rows +0..+2) ----
      {
        const float q0 = __builtin_fabsf(d[0]) - bb0[a];
        const float q1 = __builtin_fabsf(d[1]) - bb1[a];
        const float q2 = __builtin_fabsf(d[2]) - bb2[a];
        const float u0 = fmaxf(q0, SDF_EPS);
        const float u1 = fmaxf(q1, SDF_EPS);
        const float u2 = fmaxf(q2, SDF_EPS);
        const float up = __builtin_sqrtf(u0 * u0 + u1 * u1 + u2 * u2);
        const float dn = fminf(fmaxf(fmaxf(q0, q1), q2), -SDF_EPS);
        acc += __builtin_exp2f((up + dn) * EXP2_SCALE);
      }
      // ---- primitive #2 of this lane (rows +4..+6) ----
      {
        const float q0 = __builtin_fabsf(d[4]) - bb3[a];
        const float q1 = __builtin_fabsf(d[5]) - bb4[a];
        const float q2 = __builtin_fabsf(d[6]) - bb5[a];
        const float u0 = fmaxf(q0, SDF_EPS);
        const float u1 = fmaxf(q1, SDF_EPS);
        const float u2 = fmaxf(q2, SDF_EPS);
        const float up = __builtin_sqrtf(u0 * u0 + u1 * u1 + u2 * u2);
        const float dn = fminf(fmaxf(fmaxf(q0, q1), q2), -SDF_EPS);
        acc += __builtin_exp2f((up + dn) * EXP2_SCALE);
      }
    }

    // Combine the two lane-halves handling the same point (SWAPX16).
    const float other = __int_as_float(
        __builtin_amdgcn_ds_swizzle(__float_as_int(acc), 0x401F));
    const float tot = acc + other;

    if (!hi) {
      const int o = base + m;
      if (o < P)
        out[o] = -__builtin_log2f(fmaxf(tot, SDF_LOGCLAMP)) * LOG_SCALE;
    }
  }
}

extern "C" void kernel_launch(void* const* d_in, const int* in_sizes, int n_in,
                              void* d_out, int out_size, void* d_ws, size_t ws_size,
                              hipStream_t stream) {
  const float* p       = (const float*)d_in[0];  // (P,3)
  const float* centers = (const float*)d_in[1];  // (32,3)
  const float* bhalf   = (const float*)d_in[2];  // (32,3)
  const float* tfs     = (const float*)d_in[3];  // (32,3,3)
  float* out = (float*)d_out;

  const int P = in_sizes[0] / 3;
  const int numTiles = (P + 15) / 16;

  const int threads = 256;                 // 8 wave32 per block
  const int wavesPerBlock = threads / 32;
  int blocks = (numTiles + wavesPerBlock - 1) / wavesPerBlock;
  if (blocks > 2048) blocks = 2048;        // grid-stride covers the rest
  if (blocks < 1) blocks = 1;

  roundbox_sdf_wmma<<<blocks, threads, 0, stream>>>(p, centers, bhalf, tfs,
                                                    out, P, numTiles);
}